// URA_71674414235885
// MI455X (gfx1250) — compile-verified
//
#include <hip/hip_runtime.h>
#include <math.h>

#define B_    16
#define C_    64
#define H_    192
#define HW_   (192*192)          // 36864
#define BHW_  (B_*HW_)           // 589824
#define N1_   ((size_t)B_*C_*HW_) // 37748736

typedef __attribute__((ext_vector_type(2))) float v2f;
typedef __attribute__((ext_vector_type(8))) float v8f;

// D = A(16x4) * B(4x16) + C, full fp32 WMMA
__device__ __forceinline__ v8f wmma4(v2f a, v2f b, v8f c) {
    return __builtin_amdgcn_wmma_f32_16x16x4_f32(false, a, false, b, (short)0, c,
                                                 false, false);
}

// A-fragment (and transposed-B fragment): element [row0+lane%16][k0 + 2*(lane/16) + {0,1}]
__device__ __forceinline__ v2f ldA(const float* buf, int row0, int stride, int k0, int lane) {
    int lm = lane & 15;
    int ko = k0 + ((lane >> 4) << 1);
    const float* p = buf + (row0 + lm) * stride + ko;
    v2f r; r.x = p[0]; r.y = p[1];
    return r;
}

// B-fragment for row-major KxN buffer: B[k][n] at buf[k*stride + n]
__device__ __forceinline__ v2f ldBkn(const float* buf, int k0, int n0, int stride, int lane) {
    int lm = lane & 15;
    int kb = k0 + ((lane >> 4) << 1);
    v2f r; r.x = buf[kb * stride + n0 + lm];
    r.y = buf[(kb + 1) * stride + n0 + lm];
    return r;
}

// ---------------------------------------------------------------- cg pipeline
__global__ void k_init_max(unsigned int* p) { *p = 0u; }

__global__ __launch_bounds__(256)
void k_cg_blur(const float* __restrict__ map_s, const float* __restrict__ gauss,
               float* __restrict__ cgout, unsigned int* __restrict__ gmax)
{
    __shared__ float s_u[22 * 23];
    __shared__ float s_g[49];
    __shared__ float s_red[256];
    const int bx = blockIdx.x, by = blockIdx.y, b = blockIdx.z;
    const int t = threadIdx.x;

    if (t < 49) s_g[t] = gauss[t] * 4.0f;

    // stage upsample(sigmoid)->fold values for 22x22 halo tile (reflect at borders)
    for (int f = t; f < 22 * 22; f += 256) {
        int uy = f / 22, ux = f % 22;
        int gy = by * 16 + uy - 3;
        int gx = bx * 16 + ux - 3;
        if (gy < 0) gy = -gy; else if (gy > 191) gy = 382 - gy;
        if (gx < 0) gx = -gx; else if (gx > 191) gx = 382 - gx;
        // bilinear, half-pixel centers, 4x upsample, edge clamp
        float sy = gy * 0.25f - 0.375f;
        float sx = gx * 0.25f - 0.375f;
        int y0 = (int)floorf(sy); float wy = sy - (float)y0;
        int x0 = (int)floorf(sx); float wx = sx - (float)x0;
        int y0c = min(47, max(0, y0)),     x0c = min(47, max(0, x0));
        int y1c = min(47, max(0, y0 + 1)), x1c = min(47, max(0, x0 + 1));
        const float* mp = map_s + b * 48 * 48;
        float v00 = mp[y0c * 48 + x0c], v01 = mp[y0c * 48 + x1c];
        float v10 = mp[y1c * 48 + x0c], v11 = mp[y1c * 48 + x1c];
        float v = (1.f - wy) * ((1.f - wx) * v00 + wx * v01)
                +        wy  * ((1.f - wx) * v10 + wx * v11);
        float s = 1.0f / (1.0f + expf(-v));
        s_u[uy * 23 + ux] = 0.5f - fabsf(s - 0.5f);
    }
    __syncthreads();

    const int ty = t >> 4, tx = t & 15;
    float sum = 0.f;
#pragma unroll
    for (int dy = 0; dy < 7; dy++)
#pragma unroll
        for (int dx = 0; dx < 7; dx++)
            sum += s_u[(ty + dy) * 23 + (tx + dx)] * s_g[dy * 7 + dx];

    cgout[b * HW_ + (by * 16 + ty) * 192 + (bx * 16 + tx)] = sum;

    s_red[t] = sum;
    __syncthreads();
    for (int off = 128; off > 0; off >>= 1) {
        if (t < off) s_red[t] = fmaxf(s_red[t], s_red[t + off]);
        __syncthreads();
    }
    if (t == 0) atomicMax(gmax, __float_as_uint(s_red[0])); // cg >= 0 -> bits monotone
}

__global__ __launch_bounds__(256)
void k_cg_norm(float* __restrict__ cg, const unsigned int* __restrict__ gmax)
{
    int i = blockIdx.x * 256 + threadIdx.x;
    float mx = __uint_as_float(*gmax);
    cg[i] = cg[i] / mx;
}

// --------------------------------------------------------------- attention
#define XS 65       // token buffer row stride (144 x 64 padded)
#define AS 149      // attn buffer row stride (144 x 144 padded)

__global__ __launch_bounds__(256)
void k_attn(const float* __restrict__ x, const float* __restrict__ l,
            const float* __restrict__ qw, const float* __restrict__ qb,
            const float* __restrict__ kw, const float* __restrict__ kb,
            const float* __restrict__ vw, const float* __restrict__ vb,
            const float* __restrict__ cg, float* __restrict__ xr)
{
    extern __shared__ char smem[];
    float* uni  = (float*)smem;            // 85,824 B region
    float* s_x  = uni;                     // 144 x XS  (phase 0/A only)
    float* s_c  = uni + 144 * XS;          // 144 x XS  (phase 0/A only)
    float* s_at = uni;                     // 144 x AS  (overlays after A)
    float* s_q  = (float*)(smem + 85824);  // 144 x XS
    float* s_k  = s_q + 144 * XS;
    float* s_v  = s_k + 144 * XS;
    float* s_cg = s_v + 144 * XS;          // 144

    const int g  = blockIdx.x;
    const int b  = g >> 8;         // batch
    const int gg = g & 255;        // group within batch (faithful flat grouping)
    const int t    = threadIdx.x;
    const int lane = t & 31;
    const int wid  = t >> 5;
    const int lm = lane & 15, hi = lane >> 4;

    // ---- stage tokens. Flat group element F in [0,9216): token n = F%144,
    // "channel" d = F/144; image coords of F (single image channel per group):
    const int cch = gg >> 2;
    for (int f = t; f < 9216; f += 256) {
        int n = f % 144;
        int d = f / 144;
        int hw = ((gg & 3) << 2) + f / 2304;
        int R  = f % 2304;
        int ww = R / 144;
        int R2 = R % 144;
        int yy = hw * 12 + R2 / 12;
        int xx = ww * 12 + R2 % 12;
        int gi = ((b * 64 + cch) * 192 + yy) * 192 + xx;
        s_x[n * XS + d] = x[gi];
        s_c[n * XS + d] = l[gi];
    }
    if (t < 144) {  // cg windows ARE true spatial 12x12 windows (C==1)
        int hwc = gg >> 4, wwc = gg & 15;
        int yy = hwc * 12 + t / 12;
        int xx = wwc * 12 + t % 12;
        s_cg[t] = cg[b * HW_ + yy * 192 + xx];
    }
    __syncthreads();

    // ---- phase A: Q = Xw*QWt + qb ; K,V from Cw.  108 (3*9*4) WMMA tile jobs.
    for (int job = wid; job < 108; job += 8) {
        int which = job / 36;
        int rem   = job % 36;
        int mt = rem >> 2, nt = rem & 3;
        const float* src = (which == 0) ? s_x : s_c;
        const float* w   = (which == 0) ? qw : (which == 1) ? kw : vw;
        const float* bb  = (which == 0) ? qb : (which == 1) ? kb : vb;
        float*       dst = (which == 0) ? s_q : (which == 1) ? s_k : s_v;
        float bias = bb[nt * 16 + lm];
        v8f acc;
#pragma unroll
        for (int r = 0; r < 8; r++) acc[r] = bias;
        for (int k0 = 0; k0 < 64; k0 += 4) {
            v2f a  = ldA(src, mt * 16, XS, k0, lane);
            v2f bf = ldA(w,   nt * 16, 64, k0, lane);   // B[k][n] = w[n][k]
            acc = wmma4(a, bf, acc);
        }
#pragma unroll
        for (int r = 0; r < 8; r++) {
            int m = mt * 16 + r + hi * 8;
            dst[m * XS + nt * 16 + lm] = acc[r];
        }
    }
    __syncthreads();

    // ---- phase B: attn = Q*K^T * scale + log(cg outer + 0.5). 81 tiles.
    for (int job = wid; job < 81; job += 8) {
        int mt = job / 9, nt = job % 9;
        v8f acc;
#pragma unroll
        for (int r = 0; r < 8; r++) acc[r] = 0.f;
        for (int k0 = 0; k0 < 64; k0 += 4) {
            v2f a  = ldA(s_q, mt * 16, XS, k0, lane);
            v2f bf = ldA(s_k, nt * 16, XS, k0, lane);   // B[k][n] = K[n][k]
            acc = wmma4(a, bf, acc);
        }
#pragma unroll
        for (int r = 0; r < 8; r++) {
            int qr = mt * 16 + r + hi * 8;
            int kc = nt * 16 + lm;
            float bias = s_cg[qr] * s_cg[kc] + 0.5f;
            s_at[qr * AS + kc] = acc[r] * 0.125f + logf(bias);
        }
    }
    __syncthreads();

    // ---- softmax over keys, one row per thread
    if (t < 144) {
        float* row = s_at + t * AS;
        float m = row[0];
        for (int i = 1; i < 144; i++) m = fmaxf(m, row[i]);
        float s = 0.f;
        for (int i = 0; i < 144; i++) { float e = expf(row[i] - m); row[i] = e; s += e; }
        float inv = 1.0f / s;
        for (int i = 0; i < 144; i++) row[i] *= inv;
    }
    __syncthreads();

    // ---- phase C: O = attn * V, scatter with faithful flat reshape + reverse
    for (int job = wid; job < 36; job += 8) {
        int mt = job >> 2, nt = job & 3;
        v8f acc;
#pragma unroll
        for (int r = 0; r < 8; r++) acc[r] = 0.f;
        for (int k0 = 0; k0 < 144; k0 += 4) {
            v2f a  = ldA (s_at, mt * 16, AS, k0, lane);
            v2f bf = ldBkn(s_v, k0, nt * 16, XS, lane);
            acc = wmma4(a, bf, acc);
        }
#pragma unroll
        for (int r = 0; r < 8; r++) {
            int n = mt * 16 + r + hi * 8;
            int d = nt * 16 + lm;
            int F = gg * 9216 + n * 64 + d;  // flat within batch image
            int cnew = F / 36864;
            int R  = F % 36864;
            int hw = R / 2304;
            int R2 = R % 2304;
            int ww = R2 / 144;
            int R3 = R2 % 144;
            int yy = hw * 12 + R3 / 12;
            int xx = ww * 12 + R3 % 12;
            xr[((b * 64 + cnew) * 192 + yy) * 192 + xx] = acc[r];
        }
    }
}

// --------------------------------------------------- 3x3 conv (implicit GEMM)
__global__ __launch_bounds__(256)
void k_conv3x3(const float* __restrict__ src, const float* __restrict__ add,
               const float* __restrict__ w, const float* __restrict__ bias,
               float* __restrict__ dst)
{
    extern __shared__ float s_in[];   // [18][18][65] = 84,240 B
    const int bx = blockIdx.x, by = blockIdx.y, b = blockIdx.z;
    const int t = threadIdx.x;
    const int lane = t & 31, wid = t >> 5;
    const int lm = lane & 15, hi = lane >> 4;

    // stage 18x18 halo tile, 64 channels, optional residual add, zero pad
    for (int f = t; f < 18 * 18 * 64; f += 256) {
        int ci = f / 324;
        int p  = f % 324;
        int yy = p / 18, xx = p % 18;
        int gy = by * 16 + yy - 1;
        int gx = bx * 16 + xx - 1;
        float v = 0.f;
        if (gy >= 0 && gy < 192 && gx >= 0 && gx < 192) {
            int gi = ((b * 64 + ci) * 192 + gy) * 192 + gx;
            v = src[gi];
            if (add) v += add[gi];
        }
        s_in[(yy * 18 + xx) * 65 + ci] = v;
    }
    __syncthreads();

    // 16 pixel-row M-tiles x 4 out-channel N-tiles = 64 jobs
    for (int job = wid; job < 64; job += 8) {
        int mt = job >> 2;        // pixel row in tile
        int nt = job & 3;         // out-channel tile
        float bv = bias[nt * 16 + lm];
        v8f acc;
#pragma unroll
        for (int r = 0; r < 8; r++) acc[r] = bv;
#pragma unroll
        for (int dy = 0; dy < 3; dy++)
#pragma unroll
            for (int dx = 0; dx < 3; dx++) {
                int tap = dy * 3 + dx;
                for (int k0 = 0; k0 < 64; k0 += 4) {
                    int ko = k0 + (hi << 1);
                    v2f a;
                    {
                        int ba = ((mt + dy) * 18 + (lm + dx)) * 65 + ko;
                        a.x = s_in[ba]; a.y = s_in[ba + 1];
                    }
                    v2f bf;
                    {
                        int wi = ((nt * 16 + lm) * 64 + ko) * 9 + tap; // w[co][ci][tap]
                        bf.x = w[wi]; bf.y = w[wi + 9];
                    }
                    acc = wmma4(a, bf, acc);
                }
            }
#pragma unroll
        for (int r = 0; r < 8; r++) {
            int px = r + hi * 8;
            int co = nt * 16 + lm;
            dst[((b * 64 + co) * 192 + (by * 16 + mt)) * 192 + (bx * 16 + px)]
                = fmaxf(acc[r], 0.f);
        }
    }
}

// ---------------------------------------------------------------- 1x1 conv
__global__ __launch_bounds__(256)
void k_conv1x1(const float* __restrict__ x2, const float* __restrict__ w4,
               const float* __restrict__ b4, float* __restrict__ out)
{
    int p = blockIdx.x * 256 + threadIdx.x;   // [0, BHW)
    int b = p / HW_, pp = p % HW_;
    float s = b4[0];
    const float* base = x2 + (size_t)b * 64 * HW_ + pp;
#pragma unroll
    for (int ci = 0; ci < 64; ci++) s += base[ci * HW_] * w4[ci];
    out[p] = s;
}

// ---------------------------------------------------------------- launcher
extern "C" void kernel_launch(void* const* d_in, const int* in_sizes, int n_in,
                              void* d_out, int out_size, void* d_ws, size_t ws_size,
                              hipStream_t stream)
{
    const float* x     = (const float*)d_in[0];
    const float* l     = (const float*)d_in[1];
    const float* map_s = (const float*)d_in[2];
    const float* qw = (const float*)d_in[3];
    const float* qb = (const float*)d_in[4];
    const float* kw = (const float*)d_in[5];
    const float* kb = (const float*)d_in[6];
    const float* vw = (const float*)d_in[7];
    const float* vb = (const float*)d_in[8];
    const float* w1 = (const float*)d_in[9];
    const float* b1 = (const float*)d_in[10];
    const float* w3 = (const float*)d_in[11];
    const float* b3 = (const float*)d_in[12];
    const float* w4 = (const float*)d_in[13];
    const float* b4 = (const float*)d_in[14];
    const float* gauss = (const float*)d_in[15];

    float* outp   = (float*)d_out;
    float* x2buf  = outp;                       // x2 section; also xr scratch
    float* outmap = outp + N1_;                 // (B,1,192,192)
    float* cgbuf  = outmap + BHW_;              // (B,1,192,192)

    unsigned int* gmax = (unsigned int*)d_ws;
    float* tmp1 = (float*)d_ws + 16;            // conv1 output (N1_ floats)

    dim3 tiles(12, 12, 16);

    k_init_max<<<1, 1, 0, stream>>>(gmax);
    k_cg_blur<<<tiles, 256, 0, stream>>>(map_s, gauss, cgbuf, gmax);
    k_cg_norm<<<BHW_ / 256, 256, 0, stream>>>(cgbuf, gmax);

    size_t attn_lds = 85824 + 3 * (144 * XS) * sizeof(float) + 144 * sizeof(float);
    k_attn<<<4096, 256, attn_lds, stream>>>(x, l, qw, qb, kw, kb, vw, vb,
                                            cgbuf, /*xr=*/x2buf);

    size_t conv_lds = 18 * 18 * 65 * sizeof(float);
    k_conv3x3<<<tiles, 256, conv_lds, stream>>>(x2buf, nullptr, w1, b1, tmp1);
    k_conv3x3<<<tiles, 256, conv_lds, stream>>>(tmp1, x, w3, b3, x2buf);

    k_conv1x1<<<BHW_ / 256, 256, 0, stream>>>(x2buf, w4, b4, outmap);
}